// QuantumFeynmanNN_50122268344528
// MI455X (gfx1250) — compile-verified
//
#include <hip/hip_runtime.h>
#include <math.h>

// Problem constants (from reference)
#define SS     2048
#define NB     4
#define NBINS  8192          // 32KB LDS histogram, bin width 4/8192 on d2 in [0,4]
#define RANKV  1746929ull    // r-th smallest d2 == k-th smallest prop (k=2447376)
#define EPSV   1e-8f
#define LAPC   0.001f        // DT * DIFF
#define NLOG2E 1.4426950408889634f

typedef float v2f __attribute__((ext_vector_type(2)));
typedef float v8f __attribute__((ext_vector_type(8)));

__device__ __forceinline__ float half_reduce16(float v) {
  // sum across the 16 lanes of each half-wave (xor masks < 16 stay in-half)
  #pragma unroll
  for (int m = 1; m < 16; m <<= 1) v += __shfl_xor(v, m, 32);
  return v;
}

// ---------------------------------------------------------------- gather
__global__ void k_gather(const float* __restrict__ emb, const int* __restrict__ tok,
                         float* __restrict__ psi0, float* __restrict__ psiCur,
                         float* __restrict__ norms0, float* __restrict__ sq0) {
  int i = blockIdx.x * blockDim.x + threadIdx.x;
  if (i >= NB * SS) return;
  int t = tok[i];
  float x = emb[2 * t], y = emb[2 * t + 1];
  psi0[2 * i] = x;  psi0[2 * i + 1] = y;
  psiCur[2 * i] = x; psiCur[2 * i + 1] = y;
  float s = x * x + y * y;
  sq0[i] = s;
  norms0[i] = sqrtf(s);
}

__global__ void k_zero_u(unsigned* __restrict__ p, int n) {
  int i = blockIdx.x * blockDim.x + threadIdx.x;
  if (i < n) p[i] = 0u;
}

// ---------------------------------------------------------------- histogram of d2
// grid (NB*16, 4), block 256 = 8 waves. Each wave owns one i-tile (16 rows),
// sweeps 32 j-tiles. WMMA f32 16x16x4 computes the rank-2 Gram tile.
__global__ void __launch_bounds__(256)
k_hist(const float* __restrict__ base, const float* __restrict__ sqb,
       unsigned* __restrict__ ghist) {
  __shared__ unsigned sh[NBINS];
  for (int i = threadIdx.x; i < NBINS; i += 256) sh[i] = 0u;
  __syncthreads();

  int b = blockIdx.x >> 4, grp = blockIdx.x & 15;
  int wave = threadIdx.x >> 5, lane = threadIdx.x & 31;
  int lj = lane & 15, half = lane >> 4;
  int i0 = (grp * 8 + wave) * 16;
  const float* P  = base + (size_t)b * SS * 2;
  const float* SQ = sqb  + (size_t)b * SS;

  // A operand (16x4 f32): lanes 0-15 carry K=0,1 = (x_i, y_i); K=2,3 zero.
  v2f a;
  a.x = (half == 0) ? P[2 * (i0 + lj)]     : 0.0f;
  a.y = (half == 0) ? P[2 * (i0 + lj) + 1] : 0.0f;
  float sqi[8];
  #pragma unroll
  for (int v = 0; v < 8; ++v) sqi[v] = SQ[i0 + v + 8 * half];

  int j0 = blockIdx.y * 512;
  for (int jt = 0; jt < 32; ++jt) {
    int jb = j0 + jt * 16;
    float xj = P[2 * (jb + lj)], yj = P[2 * (jb + lj) + 1];
    v2f bb;
    bb.x = (half == 0) ? xj : 0.0f;   // B (4x16): rows K=0,1 = (x_j, y_j)
    bb.y = (half == 0) ? yj : 0.0f;
    float sqj = SQ[jb + lj];
    v8f c = {};
    v8f g = __builtin_amdgcn_wmma_f32_16x16x4_f32(false, a, false, bb,
                                                  (short)0, c, false, false);
    #pragma unroll
    for (int v = 0; v < 8; ++v) {
      float d2 = __builtin_fmaf(-2.0f, g[v], sqi[v] + sqj);
      int bin = (int)(d2 * (NBINS / 4.0f));
      bin = bin < 0 ? 0 : (bin > NBINS - 1 ? NBINS - 1 : bin);
      atomicAdd(&sh[bin], 1u);                       // ds_add_u32
    }
  }
  __syncthreads();
  unsigned* gh = ghist + (size_t)b * NBINS;
  for (int i = threadIdx.x; i < NBINS; i += 256)
    if (sh[i]) atomicAdd(&gh[i], sh[i]);
}

// ---------------------------------------------------------------- rank -> threshold
__global__ void k_thresh(const unsigned* __restrict__ hist, float* __restrict__ T) {
  int b = blockIdx.x;
  const unsigned* h = hist + (size_t)b * NBINS;
  __shared__ unsigned part[256];
  unsigned s = 0;
  const int chunk = NBINS / 256;
  for (int i = 0; i < chunk; ++i) s += h[threadIdx.x * chunk + i];
  part[threadIdx.x] = s;
  __syncthreads();
  if (threadIdx.x == 0) {
    unsigned long long cum = 0;
    int seg = 0;
    for (; seg < 256; ++seg) {
      if (cum + part[seg] >= RANKV) break;
      cum += part[seg];
    }
    int binIdx = NBINS - 1;
    if (seg < 256) {
      for (int i = 0; i < chunk; ++i) {
        int bi = seg * chunk + i;
        cum += h[bi];
        if (cum >= RANKV) { binIdx = bi; break; }
      }
    }
    T[b] = (binIdx + 1) * (4.0f / NBINS);   // keep d2 <= T
  }
}

// ---------------------------------------------------------------- masked matvec
// out_i accumulators: sum_j p_ij * (vx, vy, 1). grid (NB*16, 4): j-sweep split 4x,
// each split writes its own slot (deterministic, no float atomics).
__global__ void __launch_bounds__(256)
k_apply(const float* __restrict__ base, const float* __restrict__ sqb,
        const float* __restrict__ Tv, const float* __restrict__ vec,
        const float* __restrict__ sigp, float* __restrict__ accBuf) {
  int b = blockIdx.x >> 4, grp = blockIdx.x & 15;
  int wave = threadIdx.x >> 5, lane = threadIdx.x & 31;
  int lj = lane & 15, half = lane >> 4;
  int i0 = (grp * 8 + wave) * 16;
  const float* P  = base + (size_t)b * SS * 2;
  const float* V  = vec  + (size_t)b * SS * 2;
  const float* SQ = sqb  + (size_t)b * SS;
  float T  = Tv[b];
  float nc = -NLOG2E / (2.0f * sigp[0]);   // exp(-d2/2s^2) = exp2(nc * d2)

  v2f a;
  a.x = (half == 0) ? P[2 * (i0 + lj)]     : 0.0f;
  a.y = (half == 0) ? P[2 * (i0 + lj) + 1] : 0.0f;
  float sqi[8];
  #pragma unroll
  for (int v = 0; v < 8; ++v) sqi[v] = SQ[i0 + v + 8 * half];

  float accX[8], accY[8], accS[8];
  #pragma unroll
  for (int v = 0; v < 8; ++v) { accX[v] = 0.f; accY[v] = 0.f; accS[v] = 0.f; }

  int j0 = blockIdx.y * 512;
  for (int jt = 0; jt < 32; ++jt) {
    int jb = j0 + jt * 16;
    float xj = P[2 * (jb + lj)], yj = P[2 * (jb + lj) + 1];
    float vx = V[2 * (jb + lj)], vy = V[2 * (jb + lj) + 1];
    float sqj = SQ[jb + lj];
    v2f bb;
    bb.x = (half == 0) ? xj : 0.0f;
    bb.y = (half == 0) ? yj : 0.0f;
    v8f c = {};
    v8f g = __builtin_amdgcn_wmma_f32_16x16x4_f32(false, a, false, bb,
                                                  (short)0, c, false, false);
    #pragma unroll
    for (int v = 0; v < 8; ++v) {
      float d2 = __builtin_fmaf(-2.0f, g[v], sqi[v] + sqj);
      float p  = __builtin_amdgcn_exp2f(d2 * nc);    // v_exp_f32
      p = (d2 <= T) ? p : 0.0f;                      // sparsity mask (diag d2=0 kept)
      accX[v] = __builtin_fmaf(p, vx, accX[v]);
      accY[v] = __builtin_fmaf(p, vy, accY[v]);
      accS[v] += p;
    }
  }
  #pragma unroll
  for (int v = 0; v < 8; ++v) {
    accX[v] = half_reduce16(accX[v]);
    accY[v] = half_reduce16(accY[v]);
    accS[v] = half_reduce16(accS[v]);
  }
  if (lj == 0) {
    #pragma unroll
    for (int v = 0; v < 8; ++v) {
      int i = i0 + 8 * half + v;
      float* dst = accBuf + (((size_t)b * SS + i) * 4 + blockIdx.y) * 3;
      dst[0] = accX[v]; dst[1] = accY[v]; dst[2] = accS[v];
    }
  }
}

// kOut = DT*DIFF * (rownorm(matvec) - vec); fixed-order sum over 4 splits.
__global__ void k_finish(float* __restrict__ kOut, const float* __restrict__ accBuf,
                         const float* __restrict__ vec) {
  int i = blockIdx.x * blockDim.x + threadIdx.x;
  if (i >= NB * SS) return;
  float ax = 0.f, ay = 0.f, as = 0.f;
  const float* src = accBuf + (size_t)i * 12;
  #pragma unroll
  for (int s = 0; s < 4; ++s) { ax += src[3*s]; ay += src[3*s+1]; as += src[3*s+2]; }
  float inv = 1.0f / (as + EPSV);
  kOut[2*i]     = LAPC * (ax * inv - vec[2*i]);
  kOut[2*i + 1] = LAPC * (ay * inv - vec[2*i + 1]);
}

// ---------------------------------------------------------------- small elementwise
__global__ void k_acc(float* __restrict__ dst, const float* __restrict__ src,
                      float w, int init) {
  int i = blockIdx.x * blockDim.x + threadIdx.x;
  if (i >= NB * SS * 2) return;
  dst[i] = init ? w * src[i] : __builtin_fmaf(w, src[i], dst[i]);
}

__global__ void k_axpy_renorm(float* __restrict__ dstPsi, float* __restrict__ dstSq,
                              const float* __restrict__ cur, const float* __restrict__ kk,
                              float alpha, const float* __restrict__ norms0) {
  int i = blockIdx.x * blockDim.x + threadIdx.x;
  if (i >= NB * SS) return;
  float x = __builtin_fmaf(alpha, kk[2*i],     cur[2*i]);
  float y = __builtin_fmaf(alpha, kk[2*i + 1], cur[2*i + 1]);
  float n = sqrtf(x * x + y * y);
  float sc = norms0[i] / (n + EPSV);
  x *= sc; y *= sc;
  dstPsi[2*i] = x; dstPsi[2*i + 1] = y;
  dstSq[i] = x * x + y * y;
}

__global__ void k_combine(float* __restrict__ dst, const float* __restrict__ cur,
                          const float* __restrict__ kacc, const float* __restrict__ norms0) {
  int i = blockIdx.x * blockDim.x + threadIdx.x;
  if (i >= NB * SS) return;
  float x = __builtin_fmaf(1.0f / 6.0f, kacc[2*i],     cur[2*i]);
  float y = __builtin_fmaf(1.0f / 6.0f, kacc[2*i + 1], cur[2*i + 1]);
  float n = sqrtf(x * x + y * y);
  float sc = norms0[i] / (n + EPSV);
  dst[2*i] = x * sc; dst[2*i + 1] = y * sc;
}

// ---------------------------------------------------------------- host orchestration
extern "C" void kernel_launch(void* const* d_in, const int* in_sizes, int n_in,
                              void* d_out, int out_size, void* d_ws, size_t ws_size,
                              hipStream_t stream) {
  (void)in_sizes; (void)n_in; (void)out_size; (void)ws_size;
  const float* emb = (const float*)d_in[0];
  const float* sig = (const float*)d_in[1];
  const int*   tok = (const int*)d_in[2];
  float* out = (float*)d_out;

  float* w = (float*)d_ws;
  size_t o = 0;
  float* psi0   = w + o; o += 2 * NB * SS;
  float* psiCur = w + o; o += 2 * NB * SS;
  float* psiT   = w + o; o += 2 * NB * SS;
  float* kOut   = w + o; o += 2 * NB * SS;
  float* kAcc   = w + o; o += 2 * NB * SS;
  float* norms0 = w + o; o += NB * SS;
  float* sq0    = w + o; o += NB * SS;
  float* sqT    = w + o; o += NB * SS;
  float* accBuf = w + o; o += 12 * NB * SS;
  float* T0     = w + o; o += 16;
  float* Tk     = w + o; o += 16;
  unsigned* hist = (unsigned*)(w + o); o += (size_t)NB * NBINS;

  dim3 blk(256);
  dim3 gE((NB * SS + 255) / 256);
  dim3 gE2((NB * SS * 2 + 255) / 256);
  dim3 gT(NB * 16, 4);
  dim3 gZ((NB * NBINS + 255) / 256);

  auto histo = [&](const float* base, const float* sq, float* T) {
    k_zero_u<<<gZ, blk, 0, stream>>>(hist, NB * NBINS);
    k_hist<<<gT, blk, 0, stream>>>(base, sq, hist);
    k_thresh<<<NB, blk, 0, stream>>>(hist, T);
  };
  auto apply = [&](const float* base, const float* sq, const float* T,
                   const float* vec) {
    k_apply<<<gT, blk, 0, stream>>>(base, sq, T, vec, sig, accBuf);
    k_finish<<<gE, blk, 0, stream>>>(kOut, accBuf, vec);
  };

  k_gather<<<gE, blk, 0, stream>>>(emb, tok, psi0, psiCur, norms0, sq0);
  histo(psi0, sq0, T0);   // threshold of the initial propagator (reused by every k1)

  for (int step = 0; step < 3; ++step) {
    apply(psi0, sq0, T0, psiCur);                                        // k1
    k_acc<<<gE2, blk, 0, stream>>>(kAcc, kOut, 1.0f, 1);
    k_axpy_renorm<<<gE, blk, 0, stream>>>(psiT, sqT, psiCur, kOut, 0.5f, norms0);
    histo(psiT, sqT, Tk);
    apply(psiT, sqT, Tk, psiT);                                          // k2
    k_acc<<<gE2, blk, 0, stream>>>(kAcc, kOut, 2.0f, 0);
    k_axpy_renorm<<<gE, blk, 0, stream>>>(psiT, sqT, psiCur, kOut, 0.5f, norms0);
    histo(psiT, sqT, Tk);
    apply(psiT, sqT, Tk, psiT);                                          // k3
    k_acc<<<gE2, blk, 0, stream>>>(kAcc, kOut, 2.0f, 0);
    k_axpy_renorm<<<gE, blk, 0, stream>>>(psiT, sqT, psiCur, kOut, 1.0f, norms0);
    histo(psiT, sqT, Tk);
    apply(psiT, sqT, Tk, psiT);                                          // k4
    k_acc<<<gE2, blk, 0, stream>>>(kAcc, kOut, 1.0f, 0);
    float* dst = (step == 2) ? out : psiCur;
    k_combine<<<gE, blk, 0, stream>>>(dst, psiCur, kAcc, norms0);
  }
}